// PansharpSimulator_41781441855987
// MI455X (gfx1250) — compile-verified
//
#include <hip/hip_runtime.h>

typedef float v2f __attribute__((ext_vector_type(2)));
typedef float v8f __attribute__((ext_vector_type(8)));

#define H_IN 1024
#define W_IN 1024
#define NB 4
#define NC 4
#define NM 41      // MTF kernel size
#define KC 44      // combined (MTF conv box4) kernel size
#define HL 256     // lrms spatial size
#define TIN 104    // input tile edge for one 16x16 lrms tile: 4*15+43+1
#define TS  104    // LDS row stride (even -> 8B-aligned b64 frag loads)

// ---------------------------------------------------------------------------
// Kernel 1: build combined stride-4 kernel  Kc[u,v] = (1/16) * sum_{dy,dx<4} K[u-dy, v-dx]
// ---------------------------------------------------------------------------
__global__ void build_kc_kernel(const float* __restrict__ mtf, float* __restrict__ kc) {
    int idx = blockIdx.x * blockDim.x + threadIdx.x;
    if (idx >= NC * KC * KC) return;
    int band = idx / (KC * KC);
    int rem  = idx % (KC * KC);
    int u = rem / KC, v = rem % KC;
    const float* k = mtf + band * NM * NM;
    float s = 0.f;
    for (int dy = 0; dy < 4; ++dy) {
        int i = u - dy; if (i < 0 || i >= NM) continue;
        for (int dx = 0; dx < 4; ++dx) {
            int j = v - dx; if (j < 0 || j >= NM) continue;
            s += k[i * NM + j];
        }
    }
    kc[idx] = s * (1.0f / 16.0f);
}

// ---------------------------------------------------------------------------
// Kernel 2: pan = sum_c w[c] * x[b,c,:,:]   (float4 vectorized, pure BW)
// ---------------------------------------------------------------------------
__global__ void pan_kernel(const float* __restrict__ x, const float* __restrict__ w,
                           float* __restrict__ pan) {
    const int planeV = H_IN * W_IN / 4;
    int i = blockIdx.x * blockDim.x + threadIdx.x;
    if (i >= NB * planeV) return;
    int b = i / planeV, p = i % planeV;
    const float4* xb = (const float4*)x + (size_t)b * NC * planeV;
    float w0 = w[0], w1 = w[1], w2 = w[2], w3 = w[3];
    float4 a0 = xb[p];
    float4 a1 = xb[(size_t)planeV + p];
    float4 a2 = xb[2 * (size_t)planeV + p];
    float4 a3 = xb[3 * (size_t)planeV + p];
    float4 r;
    r.x = a0.x * w0 + a1.x * w1 + a2.x * w2 + a3.x * w3;
    r.y = a0.y * w0 + a1.y * w1 + a2.y * w2 + a3.y * w3;
    r.z = a0.z * w0 + a1.z * w1 + a2.z * w2 + a3.z * w3;
    r.w = a0.w * w0 + a1.w * w1 + a2.w * w2 + a3.w * w3;
    ((float4*)pan)[i] = r;
}

// ---------------------------------------------------------------------------
// Kernel 3: fused MTF-blur + 4x4 mean pool via f32 WMMA (Toeplitz mapping).
//   lrms[Y,X] = sum_{u,v<44} Kc[u,v] * x[clamp(4Y+u-20), clamp(4X+v-20)]
//   Per 16x16 output tile: D[m=Y, n=X] += A[m,s] * B[s,n],
//   A[m,s] = tile row 4m+u, col s;   B[s,n] = Kc[u, s-4n]  (banded Toeplitz)
//   26 s-chunks of 4, 44 u-rows -> 1144 v_wmma_f32_16x16x4_f32 per tile,
//   split over 4 waves by u-range, reduced through LDS.
//   Out-of-band B lanes read from a zero-filled LDS shadow array instead of
//   masking with a per-WMMA multiply -> inner loop is load+load+wmma only.
// ---------------------------------------------------------------------------
__global__ __launch_bounds__(128) void lrms_wmma_kernel(const float* __restrict__ x,
                                                        const float* __restrict__ kc,
                                                        float* __restrict__ lrms) {
    __shared__ __align__(16) float sIn[TIN * TS];    // 104x104 input tile (43.3 KB)
    __shared__ __align__(16) float sK[KC * KC];      // 44x44 combined kernel (7.7 KB)
    __shared__ __align__(16) float sKz[KC * KC];     // zero shadow (7.7 KB)
    __shared__ __align__(16) float sRed[3 * 32 * 8]; // partial-acc reduction (3 KB)

    int plane = blockIdx.z;               // b*NC + c
    int band  = plane & 3;
    int X0 = blockIdx.x * 16, Y0 = blockIdx.y * 16;
    const float* xin = x + (size_t)plane * (size_t)(H_IN * W_IN);
    int tid = threadIdx.x;

    for (int i = tid; i < KC * KC; i += 128) {
        sK[i]  = kc[band * KC * KC + i];
        sKz[i] = 0.0f;
    }

    int rbase = 4 * Y0 - 20, cbase = 4 * X0 - 20;   // edge-clamped tile origin
    for (int i = tid; i < TIN * TIN; i += 128) {
        int r = i / TIN, cc = i % TIN;
        int gr = min(max(rbase + r, 0), H_IN - 1);
        int gc = min(max(cbase + cc, 0), W_IN - 1);
        sIn[r * TS + cc] = xin[gr * W_IN + gc];
    }
    __syncthreads();

    int wave = tid >> 5, lane = tid & 31;
    int mrow = lane & 15;         // M for A-frag, N for B-frag (same lane group)
    int kh   = lane >> 4;         // which K-half this lane holds (ISA 16x4 layout)
    int u0 = wave * 11, u1 = u0 + 11;   // 44 u-rows split over 4 waves

    v8f acc = {0.f, 0.f, 0.f, 0.f, 0.f, 0.f, 0.f, 0.f};

    for (int cch = 0; cch < 26; ++cch) {
        int aCol = 4 * cch + 2 * kh;          // even -> aligned b64
        int v0   = aCol - 4 * mrow;           // Toeplitz band offset (even)
        bool inB = (v0 >= 0) && (v0 < KC);
        const float* aBase = &sIn[(4 * mrow) * TS + aCol];
        const float* bBase = inB ? &sK[v0] : &sKz[0];   // zero shadow: no masking
        #pragma unroll 4
        for (int u = u0; u < u1; ++u) {
            v2f a = *(const v2f*)(aBase + u * TS);
            v2f b = *(const v2f*)(bBase + u * KC);
            acc = __builtin_amdgcn_wmma_f32_16x16x4_f32(
                /*neg_a=*/false, a, /*neg_b=*/false, b,
                /*c_mod=*/(short)0, acc, /*reuse_a=*/false, /*reuse_b=*/false);
        }
    }

    if (wave > 0) {
        float* dst = &sRed[((wave - 1) * 32 + lane) * 8];
        #pragma unroll
        for (int q = 0; q < 8; ++q) dst[q] = acc[q];
    }
    __syncthreads();
    if (wave == 0) {
        #pragma unroll
        for (int q = 0; q < 8; ++q)
            acc[q] += sRed[(0 * 32 + lane) * 8 + q] + sRed[(1 * 32 + lane) * 8 + q] +
                      sRed[(2 * 32 + lane) * 8 + q];
        // D layout: lanes 0-15 -> rows 0..7, col=lane; lanes 16-31 -> rows 8..15
        int colOut  = X0 + mrow;
        int rowBase = Y0 + 8 * kh;
        float* outP = lrms + (size_t)plane * (HL * HL);
        #pragma unroll
        for (int q = 0; q < 8; ++q)
            outP[(rowBase + q) * HL + colOut] = acc[q];
    }
}

// ---------------------------------------------------------------------------
// Kernels 4/5: polyphase x2 upsample + separable 23-tap circular filter.
// Zero-interleaved kernel => even-phase outputs copy; odd-phase use the 12
// nonzero taps k23[2t]. PHASE = data parity after zero insertion
// (stage1 data at [1::2,1::2] => PHASE=1; stage2 at [::2,::2] => PHASE=0).
// ---------------------------------------------------------------------------
template <int P, int PHASE>
__global__ __launch_bounds__(256) void upsample_stage(const float* __restrict__ in,
                                                      float* __restrict__ out,
                                                      const float* __restrict__ k23) {
    const int NI  = 28;                         // cached input rows/cols (16 + 12 halo)
    const int OFF = (PHASE == 1) ? 6 : 5;       // tap base offset
    __shared__ float sI[NI][NI + 2];
    __shared__ float sT[NI][33];                // horizontal-pass result
    __shared__ float sW[12];

    int plane = blockIdx.z;
    int X0 = blockIdx.x * 32, Y0 = blockIdx.y * 32;
    int tid = threadIdx.x;
    if (tid < 12) sW[tid] = k23[2 * tid];

    int i0 = Y0 / 2 - OFF, j0 = X0 / 2 - OFF;
    const float* inP = in + (size_t)plane * P * P;
    for (int e = tid; e < NI * NI; e += 256) {
        int r = e / NI, cq = e % NI;
        int gr = i0 + r;  gr = (gr < 0) ? gr + P : (gr >= P ? gr - P : gr);   // wrap
        int gc = j0 + cq; gc = (gc < 0) ? gc + P : (gc >= P ? gc - P : gc);
        sI[r][cq] = inP[gr * P + gc];
    }
    __syncthreads();

    for (int e = tid; e < NI * 32; e += 256) {   // horizontal pass
        int r = e / 32, xo = e % 32;
        int xg = X0 + xo;
        float val;
        if ((xg & 1) == PHASE) {
            val = sI[r][((xg - PHASE) >> 1) - j0];
        } else {
            int cb = ((PHASE == 1) ? (xg >> 1) - 6 : ((xg - 1) >> 1) - 5) - j0;
            float s = 0.f;
            #pragma unroll
            for (int t = 0; t < 12; ++t) s += sW[t] * sI[r][cb + t];
            val = s;
        }
        sT[r][xo] = val;
    }
    __syncthreads();

    float* outP = out + (size_t)plane * (size_t)(2 * P) * (2 * P);
    for (int e = tid; e < 32 * 32; e += 256) {   // vertical pass
        int yo = e / 32, xo = e % 32;
        int yg = Y0 + yo;
        float val;
        if ((yg & 1) == PHASE) {
            val = sT[((yg - PHASE) >> 1) - i0][xo];
        } else {
            int ib = ((PHASE == 1) ? (yg >> 1) - 6 : ((yg - 1) >> 1) - 5) - i0;
            float s = 0.f;
            #pragma unroll
            for (int t = 0; t < 12; ++t) s += sW[t] * sT[ib + t][xo];
            val = s;
        }
        outP[(size_t)yg * (2 * P) + X0 + xo] = val;
    }
}

// ---------------------------------------------------------------------------
extern "C" void kernel_launch(void* const* d_in, const int* in_sizes, int n_in,
                              void* d_out, int out_size, void* d_ws, size_t ws_size,
                              hipStream_t stream) {
    (void)in_sizes; (void)n_in; (void)out_size; (void)ws_size;
    const float* x   = (const float*)d_in[0];   // (4,4,1024,1024)
    const float* mtf = (const float*)d_in[1];   // (4,41,41)
    const float* ik  = (const float*)d_in[2];   // (23,)
    const float* pw  = (const float*)d_in[3];   // (4,)
    // d_in[4] = ratio (always 4 in this problem)

    float* out  = (float*)d_out;
    float* pan  = out;                                        // 4*1024*1024
    float* lrms = out + (size_t)4 * 1024 * 1024;              // 4*4*256*256
    float* msup = lrms + (size_t)4 * 4 * 256 * 256;           // 4*4*1024*1024

    float* kc = (float*)d_ws;                                 // 4*44*44 floats
    float* o1 = (float*)((char*)d_ws + 32768);                // 16*512*512 floats

    build_kc_kernel<<<dim3((NC * KC * KC + 255) / 256), dim3(256), 0, stream>>>(mtf, kc);

    pan_kernel<<<dim3((NB * H_IN * W_IN / 4 + 255) / 256), dim3(256), 0, stream>>>(x, pw, pan);

    lrms_wmma_kernel<<<dim3(16, 16, 16), dim3(128), 0, stream>>>(x, kc, lrms);

    upsample_stage<256, 1><<<dim3(16, 16, 16), dim3(256), 0, stream>>>(lrms, o1, ik);
    upsample_stage<512, 0><<<dim3(32, 32, 16), dim3(256), 0, stream>>>(o1, msup, ik);
}